// GCNLayer_12086037971597
// MI455X (gfx1250) — compile-verified
//
#include <hip/hip_runtime.h>

typedef __attribute__((ext_vector_type(2))) float v2f;
typedef __attribute__((ext_vector_type(8))) float v8f;

#define GCN_N_NODES 100000
#define GCN_IN_DIM  256
#define GCN_OUT_DIM 128

// ---------------------------------------------------------------------------
// Kernel 1: H[N,128] = X[N,256] @ W[128,256]^T + b   via V_WMMA_F32_16X16X4_F32
// One wave32 per 16-row strip; 8 C-tiles (16x16) = full 16x128 strip per wave.
// A fragment layout (32-bit A 16x4, ISA 7.12.2):
//   lanes 0-15 : row m0+lane, K = k+0 (v0), k+1 (v1)
//   lanes 16-31: row m0+lane-16, K = k+2 (v0), k+3 (v1)
// B fragment layout (4x16, row striped across lanes):
//   lanes 0-15 : N = o0+lane, K = k+0 (v0), k+1 (v1)
//   lanes 16-31: N = o0+lane-16, K = k+2 (v0), k+3 (v1)
// Both are a single 8-byte-aligned float2 load per lane per K-step.
// ---------------------------------------------------------------------------
__global__ __launch_bounds__(32)
void gcn_gemm_wmma_f32(const float* __restrict__ X,
                       const float* __restrict__ W,
                       const float* __restrict__ bias,
                       float* __restrict__ H)
{
    const int mTile  = blockIdx.x;          // 0..6249 (exact, no remainder)
    const int lane   = threadIdx.x;         // 0..31
    const int lane15 = lane & 15;
    const int hi     = lane >> 4;           // 0 or 1
    const int m0     = mTile * 16;

    const float* aPtr = X + (size_t)(m0 + lane15) * GCN_IN_DIM + 2 * hi;

    const float* bPtr[8];
#pragma unroll
    for (int j = 0; j < 8; ++j)
        bPtr[j] = W + (size_t)(16 * j + lane15) * GCN_IN_DIM + 2 * hi;

    v8f acc[8];
#pragma unroll
    for (int j = 0; j < 8; ++j)
        acc[j] = (v8f){0.f, 0.f, 0.f, 0.f, 0.f, 0.f, 0.f, 0.f};

    for (int k = 0; k < GCN_IN_DIM; k += 4) {
        v2f a = *(const v2f*)(aPtr + k);     // reused across all 8 B tiles
#pragma unroll
        for (int j = 0; j < 8; ++j) {
            v2f bv = *(const v2f*)(bPtr[j] + k);
            acc[j] = __builtin_amdgcn_wmma_f32_16x16x4_f32(
                /*neg_a=*/false, a, /*neg_b=*/false, bv,
                /*c_mod=*/(short)0, acc[j],
                /*reuse_a=*/false, /*reuse_b=*/false);
        }
    }

    // C/D layout: VGPR v holds row (m0 + v + 8*hi), column lane15 of tile j.
#pragma unroll
    for (int j = 0; j < 8; ++j) {
        const float bj = bias[16 * j + lane15];
#pragma unroll
        for (int v = 0; v < 8; ++v) {
            const int row = m0 + v + 8 * hi;
            H[(size_t)row * GCN_OUT_DIM + 16 * j + lane15] = acc[j][v] + bj;
        }
    }
}

// ---------------------------------------------------------------------------
// Kernel 2: zero the output accumulator (harness poisons d_out to 0xAA).
// ---------------------------------------------------------------------------
__global__ void gcn_zero_out(float4* __restrict__ out, int n4)
{
    int i = blockIdx.x * blockDim.x + threadIdx.x;
    if (i < n4) {
        float4 z;
        z.x = 0.f; z.y = 0.f; z.z = 0.f; z.w = 0.f;
        out[i] = z;
    }
}

// ---------------------------------------------------------------------------
// Kernel 3: edge scatter.  One wave32 per edge: lane l handles floats
// [4l, 4l+4) of the 128-wide row (coalesced 512B per wave).  H rows gather
// mostly from L2 (H = 51.2MB < 192MB L2); accumulation uses non-returning
// global_atomic_add_f32 (STOREcnt-tracked, no return traffic).
// ---------------------------------------------------------------------------
__global__ __launch_bounds__(256)
void gcn_edge_scatter(const int*   __restrict__ edge_row,
                      const int*   __restrict__ edge_col,
                      const float* __restrict__ edge_val,
                      const float* __restrict__ H,
                      float*       __restrict__ out,
                      int n_edges)
{
    const int e    = (int)((blockIdx.x * (unsigned)blockDim.x + threadIdx.x) >> 5);
    const int lane = threadIdx.x & 31;
    if (e >= n_edges) return;

    const int   r = edge_row[e];
    const int   c = edge_col[e];
    const float v = edge_val[e];

    const float4 h = ((const float4*)(H + (size_t)c * GCN_OUT_DIM))[lane];
    float* orow = out + (size_t)r * GCN_OUT_DIM + lane * 4;

    unsafeAtomicAdd(orow + 0, v * h.x);
    unsafeAtomicAdd(orow + 1, v * h.y);
    unsafeAtomicAdd(orow + 2, v * h.z);
    unsafeAtomicAdd(orow + 3, v * h.w);
}

// ---------------------------------------------------------------------------
// Launch: GEMM -> zero -> scatter (stream-ordered).
// d_in order: X, edge_row, edge_col, edge_val, W, b
// ---------------------------------------------------------------------------
extern "C" void kernel_launch(void* const* d_in, const int* in_sizes, int n_in,
                              void* d_out, int out_size, void* d_ws, size_t ws_size,
                              hipStream_t stream)
{
    const float* X        = (const float*)d_in[0];
    const int*   edge_row = (const int*)d_in[1];
    const int*   edge_col = (const int*)d_in[2];
    const float* edge_val = (const float*)d_in[3];
    const float* W        = (const float*)d_in[4];
    const float* bias     = (const float*)d_in[5];
    float*       out      = (float*)d_out;
    float*       H        = (float*)d_ws;   // 100000*128*4 = 51.2 MB scratch

    const int n_edges = in_sizes[1];

    // 1) H = X @ W^T + b   (6250 waves, one 16x128 strip each)
    gcn_gemm_wmma_f32<<<GCN_N_NODES / 16, 32, 0, stream>>>(X, W, bias, H);

    // 2) out = 0
    const int n4 = GCN_N_NODES * GCN_OUT_DIM / 4;
    gcn_zero_out<<<(n4 + 255) / 256, 256, 0, stream>>>((float4*)out, n4);

    // 3) scatter-accumulate edges (8 edges per 256-thread block)
    const int nblk = (n_edges + 7) / 8;
    gcn_edge_scatter<<<nblk, 256, 0, stream>>>(edge_row, edge_col, edge_val,
                                               H, out, n_edges);
}